// dt_scaled_atnettion_19318762897737
// MI455X (gfx1250) — compile-verified
//
#include <hip/hip_runtime.h>
#include <hip/hip_bf16.h>
#include <math.h>

// ---------------- problem constants (match reference) ----------------
#define BB   4
#define NN   8192
#define CC   64
#define OUTD 64
#define KNN  36
#define PTS  (BB * NN)          // 32768 rows total

typedef __attribute__((ext_vector_type(2))) float v2f;
typedef __attribute__((ext_vector_type(8))) float v8f;

// =====================================================================
// Kernel A: KNN top-36 per query point.
// Block = 256 threads (one query each), grid = (NN/256, BB).
// Candidate xyz streamed through LDS in 2048-point chunks (24 KB);
// next chunk prefetched (global_prefetch_b8) while current is consumed.
// Top-36 kept in registers: replace current argmax, rescan (unrolled,
// constant indexing only -> stays in VGPRs, no scratch).
// =====================================================================
__global__ __launch_bounds__(256) void knn_kernel(const float* __restrict__ xyz,
                                                  int* __restrict__ idxout) {
  const int b = blockIdx.y;
  const int i = blockIdx.x * 256 + threadIdx.x;
  const float* xb = xyz + (size_t)b * NN * 3;

  const float qx = xb[i * 3 + 0];
  const float qy = xb[i * 3 + 1];
  const float qz = xb[i * 3 + 2];

  __shared__ float lx[2048], ly[2048], lz[2048];

  float bd[KNN];
  int   bi[KNN];
#pragma unroll
  for (int t = 0; t < KNN; ++t) { bd[t] = 3.4e38f; bi[t] = 0; }
  float wmax = 3.4e38f;
  int   wpos = KNN - 1;

  for (int c = 0; c < NN / 2048; ++c) {
    __syncthreads();
    for (int j = threadIdx.x; j < 2048; j += 256) {
      const int g = c * 2048 + j;
      lx[j] = xb[g * 3 + 0];
      ly[j] = xb[g * 3 + 1];
      lz[j] = xb[g * 3 + 2];
    }
    __syncthreads();

    // prefetch next chunk into cache while we chew on this one
    if (c + 1 < NN / 2048) {
      const float* nxt = xb + (size_t)((c + 1) * 2048 + threadIdx.x * 8) * 3;
      __builtin_prefetch(nxt, 0, 0);
    }

    for (int j = 0; j < 2048; ++j) {
      const float dx = qx - lx[j];
      const float dy = qy - ly[j];
      const float dz = qz - lz[j];
      const float d2 = dx * dx + dy * dy + dz * dz;
      if (d2 < wmax) {                       // rare path (~195/8192)
        const int jj = c * 2048 + j;
#pragma unroll
        for (int t = 0; t < KNN; ++t)
          if (t == wpos) { bd[t] = d2; bi[t] = jj; }
        // rescan for new worst
        wmax = bd[0]; wpos = 0;
#pragma unroll
        for (int t = 1; t < KNN; ++t)
          if (bd[t] > wmax) { wmax = bd[t]; wpos = t; }
      }
    }
  }

  int* op = idxout + ((size_t)b * NN + i) * KNN;
#pragma unroll
  for (int t = 0; t < KNN; ++t) op[t] = bi[t];
}

// =====================================================================
// Kernel B: v_feat = relu([feature|xyz] @ W_v + b_v)
// f32 WMMA 16x16x4.  Block = 128 threads (4 waves), each wave owns a
// 16-row strip and all 4 N-tiles.  K padded 67 -> 68 (17 chunks of 4).
// Weight tile stored k-pair-interleaved in LDS: Wp[k/2][n] = {W[k][n],
// W[k+1][n]} so every B fragment is ONE aligned ds_load_b64 straight
// into an even VGPR pair (no packing moves).  A fragment is contiguous
// in its row -> direct v2f load.  K-loop fully unrolled, 4 named v8f
// accumulators -> WMMA accumulates in place.
// =====================================================================
__global__ __launch_bounds__(128) void vfeat_kernel(const float* __restrict__ feat,
                                                    const float* __restrict__ xyz,
                                                    const float* __restrict__ Wv,
                                                    const float* __restrict__ bv,
                                                    float* __restrict__ vfeat) {
  __shared__ float Wp[34 * 128];  // [kpair][n*2 + {lo,hi}], k rows 67 zero-padded
  __shared__ float Al[64 * 68];   // [row][k] concat(feature, xyz, 0)

  const int tid = threadIdx.x;
  const int rowbase = blockIdx.x * 64;

  for (int i = tid; i < 34 * 64; i += 128) {
    const int kk = i / 64, n = i % 64;
    const int k0 = 2 * kk, k1 = 2 * kk + 1;
    Wp[kk * 128 + 2 * n + 0] = (k0 < 67) ? Wv[k0 * 64 + n] : 0.0f;
    Wp[kk * 128 + 2 * n + 1] = (k1 < 67) ? Wv[k1 * 64 + n] : 0.0f;
  }
  for (int i = tid; i < 64 * 68; i += 128) {
    const int r = i / 68, c = i % 68;
    const int g = rowbase + r;
    float v;
    if (c < 64)      v = feat[(size_t)g * 64 + c];
    else if (c < 67) v = xyz[(size_t)g * 3 + (c - 64)];
    else             v = 0.0f;
    Al[i] = v;
  }
  __syncthreads();

  const int wave = tid >> 5;
  const int lane = tid & 31;
  const int half = lane >> 4;     // 0: lanes 0-15 (K=k0,k0+1), 1: lanes 16-31 (K=k0+2,k0+3)
  const int lr   = lane & 15;

  const float* Arow = &Al[(wave * 16 + lr) * 68];

  v8f acc0 = {}, acc1 = {}, acc2 = {}, acc3 = {};
#pragma unroll
  for (int kt = 0; kt < 17; ++kt) {
    const int k0 = kt * 4;
    // A: elements (k0+2*half, k0+2*half+1) are contiguous & 8B aligned
    const v2f a = *(const v2f*)&Arow[k0 + 2 * half];
    // B: pair row kk = k0/2 + half, one aligned b64 load per fragment
    const float* wrow = &Wp[(2 * kt + half) * 128 + 2 * lr];
    const v2f b0 = *(const v2f*)&wrow[0];
    const v2f b1 = *(const v2f*)&wrow[32];
    const v2f b2 = *(const v2f*)&wrow[64];
    const v2f b3 = *(const v2f*)&wrow[96];
    acc0 = __builtin_amdgcn_wmma_f32_16x16x4_f32(false, a, false, b0, (short)0, acc0, false, false);
    acc1 = __builtin_amdgcn_wmma_f32_16x16x4_f32(false, a, false, b1, (short)0, acc1, false, false);
    acc2 = __builtin_amdgcn_wmma_f32_16x16x4_f32(false, a, false, b2, (short)0, acc2, false, false);
    acc3 = __builtin_amdgcn_wmma_f32_16x16x4_f32(false, a, false, b3, (short)0, acc3, false, false);
  }

  const int rtile = rowbase + wave * 16;
#pragma unroll
  for (int nt = 0; nt < 4; ++nt) {
    const v8f acc = (nt == 0) ? acc0 : (nt == 1) ? acc1 : (nt == 2) ? acc2 : acc3;
    const int col  = nt * 16 + lr;
    const float bs = bv[col];
#pragma unroll
    for (int v = 0; v < 8; ++v) {
      const int row = rtile + v + 8 * half;   // D layout: VGPR v -> M=v (+8 for hi half)
      const float x = acc[v] + bs;
      vfeat[(size_t)row * 64 + col] = fmaxf(x, 0.0f);
    }
  }
}

// =====================================================================
// Kernel C: attention.  One wave per point; lane owns channels
// {lane, lane+32}.  36 logits via wave32 xor-butterfly reduce (result
// broadcast to all lanes -> logits live in registers, no LDS).
// =====================================================================
__global__ __launch_bounds__(256) void attn_kernel(const float* __restrict__ feat,
                                                   const float* __restrict__ vfeat,
                                                   const int* __restrict__ idxbuf,
                                                   float* __restrict__ attn) {
  const int lane = threadIdx.x & 31;
  const int p    = blockIdx.x * 8 + (threadIdx.x >> 5);
  const int b    = p >> 13;                  // p / NN
  const size_t base = (size_t)p * 64;

  const float q0 = feat[base + lane];
  const float q1 = feat[base + 32 + lane];

  int idxs[KNN];
  const int* ip = idxbuf + (size_t)p * KNN;
#pragma unroll
  for (int k = 0; k < KNN; ++k) idxs[k] = ip[k];

  float w[KNN];
#pragma unroll
  for (int k = 0; k < KNN; ++k) {
    const float* f = feat + ((size_t)b * NN + idxs[k]) * 64;
    float pa = q0 * f[lane] + q1 * f[32 + lane];
#pragma unroll
    for (int off = 16; off > 0; off >>= 1) pa += __shfl_xor(pa, off, 32);
    w[k] = pa;                               // all lanes hold the full dot
  }

  float m = w[0];
#pragma unroll
  for (int k = 1; k < KNN; ++k) m = fmaxf(m, w[k]);
  float s = 0.0f;
#pragma unroll
  for (int k = 0; k < KNN; ++k) { w[k] = __expf(w[k] - m); s += w[k]; }
  const float inv = 1.0f / s;

  float a0 = 0.0f, a1 = 0.0f;
#pragma unroll
  for (int k = 0; k < KNN; ++k) {
    const float* vf = vfeat + ((size_t)b * NN + idxs[k]) * 64;
    const float wk = w[k] * inv;
    a0 += wk * vf[lane];
    a1 += wk * vf[32 + lane];
  }
  attn[base + lane]      = a0;
  attn[base + 32 + lane] = a1;
}

// =====================================================================
// Kernel D: out = attn @ W_suf + b_suf   (f32 WMMA, K=64 -> 16 chunks)
// Same k-pair-interleaved weight layout as kernel B.
// =====================================================================
__global__ __launch_bounds__(128) void suffix_kernel(const float* __restrict__ attn,
                                                     const float* __restrict__ Ws,
                                                     const float* __restrict__ bs,
                                                     float* __restrict__ out) {
  __shared__ float Wp[32 * 128];  // [kpair][n*2 + {lo,hi}]
  __shared__ float Al[64 * 64];

  const int tid = threadIdx.x;
  const int rowbase = blockIdx.x * 64;

  for (int i = tid; i < 32 * 64; i += 128) {
    const int kk = i / 64, n = i % 64;
    Wp[kk * 128 + 2 * n + 0] = Ws[(2 * kk + 0) * 64 + n];
    Wp[kk * 128 + 2 * n + 1] = Ws[(2 * kk + 1) * 64 + n];
  }
  for (int i = tid; i < 64 * 64; i += 128) {
    const int r = i / 64, c = i % 64;
    Al[i] = attn[(size_t)(rowbase + r) * 64 + c];
  }
  __syncthreads();

  const int wave = tid >> 5;
  const int lane = tid & 31;
  const int half = lane >> 4;
  const int lr   = lane & 15;

  const float* Arow = &Al[(wave * 16 + lr) * 64];

  v8f acc0 = {}, acc1 = {}, acc2 = {}, acc3 = {};
#pragma unroll
  for (int kt = 0; kt < 16; ++kt) {
    const int k0 = kt * 4;
    const v2f a = *(const v2f*)&Arow[k0 + 2 * half];
    const float* wrow = &Wp[(2 * kt + half) * 128 + 2 * lr];
    const v2f b0 = *(const v2f*)&wrow[0];
    const v2f b1 = *(const v2f*)&wrow[32];
    const v2f b2 = *(const v2f*)&wrow[64];
    const v2f b3 = *(const v2f*)&wrow[96];
    acc0 = __builtin_amdgcn_wmma_f32_16x16x4_f32(false, a, false, b0, (short)0, acc0, false, false);
    acc1 = __builtin_amdgcn_wmma_f32_16x16x4_f32(false, a, false, b1, (short)0, acc1, false, false);
    acc2 = __builtin_amdgcn_wmma_f32_16x16x4_f32(false, a, false, b2, (short)0, acc2, false, false);
    acc3 = __builtin_amdgcn_wmma_f32_16x16x4_f32(false, a, false, b3, (short)0, acc3, false, false);
  }

  const int rtile = rowbase + wave * 16;
#pragma unroll
  for (int nt = 0; nt < 4; ++nt) {
    const v8f acc = (nt == 0) ? acc0 : (nt == 1) ? acc1 : (nt == 2) ? acc2 : acc3;
    const int col   = nt * 16 + lr;
    const float bia = bs[col];
#pragma unroll
    for (int v = 0; v < 8; ++v) {
      const int row = rtile + v + 8 * half;
      out[(size_t)row * 64 + col] = acc[v] + bia;
    }
  }
}

// Tuple-return tail: reference returns (out, N); fill trailing elements.
__global__ void tail_kernel(float* __restrict__ out, int start, int total) {
  const int i = start + threadIdx.x + blockIdx.x * blockDim.x;
  if (i < total) out[i] = (float)NN;
}

// =====================================================================
extern "C" void kernel_launch(void* const* d_in, const int* in_sizes, int n_in,
                              void* d_out, int out_size, void* d_ws, size_t ws_size,
                              hipStream_t stream) {
  (void)in_sizes; (void)n_in; (void)ws_size;

  const float* feature = (const float*)d_in[0];   // (B,N,64)
  const float* xyz     = (const float*)d_in[1];   // (B,N,3)
  const float* Wv      = (const float*)d_in[2];   // (67,64)
  const float* bv      = (const float*)d_in[3];   // (64)
  const float* Wsuf    = (const float*)d_in[4];   // (64,64)
  const float* bsuf    = (const float*)d_in[5];   // (64)

  char* ws = (char*)d_ws;
  size_t off = 0;
  int* idx = (int*)(ws + off);
  off += (size_t)PTS * KNN * sizeof(int);
  off = (off + 255) & ~(size_t)255;
  float* vfeat = (float*)(ws + off);
  off += (size_t)PTS * OUTD * sizeof(float);
  off = (off + 255) & ~(size_t)255;
  float* attn = (float*)(ws + off);

  float* out = (float*)d_out;

  knn_kernel   <<<dim3(NN / 256, BB), 256, 0, stream>>>(xyz, idx);
  vfeat_kernel <<<PTS / 64, 128, 0, stream>>>(feature, xyz, Wv, bv, vfeat);
  attn_kernel  <<<PTS / 8, 256, 0, stream>>>(feature, vfeat, idx, attn);
  suffix_kernel<<<PTS / 64, 128, 0, stream>>>(attn, Wsuf, bsuf, out);

  const int body = PTS * OUTD;
  if (out_size > body) {
    const int tail = out_size - body;
    tail_kernel<<<(tail + 63) / 64, 64, 0, stream>>>(out, body, out_size);
  }
}